// DecoderLayer_68461778698665
// MI455X (gfx1250) — compile-verified
//
#include <hip/hip_runtime.h>

typedef __attribute__((ext_vector_type(2))) float v2f;
typedef __attribute__((ext_vector_type(4))) float v4f;
typedef __attribute__((ext_vector_type(8))) float v8f;

#define NPACK 16
#define NNODES 16384
#define DLAT 128
#define NGRAPH 16
#define BLOCKS_PER_PACK 32
#define NODES_PER_BLOCK (NNODES / BLOCKS_PER_PACK)   // 512
#define THREADS_A 256
#define WAVES_A (THREADS_A / 32)                     // 8 waves (wave32)
#define NODES_PER_WAVE (NODES_PER_BLOCK / WAVES_A)   // 64

// ---------------------------------------------------------------------------
// Kernel A: bandwidth-bound segment dot-sum.
// partial[block][g] = sum over nodes in this block with idx==g of nodes[n]·W[0:128]
// One node per wave-iteration: 32 lanes * float4 = full 128-dim row, coalesced.
// Scalar per-lane partial goes into LDS bins via ds_add_f32 (bank-conflict free).
// ---------------------------------------------------------------------------
__global__ __launch_bounds__(THREADS_A) void seg_dot_kernel(
    const float* __restrict__ nodes, const float* __restrict__ W,
    const int* __restrict__ idx, float* __restrict__ partials) {
  __shared__ float bins[NGRAPH * 32];
  const int tid = threadIdx.x;
  for (int i = tid; i < NGRAPH * 32; i += THREADS_A) bins[i] = 0.0f;
  __syncthreads();

  const int b     = blockIdx.x / BLOCKS_PER_PACK;
  const int chunk = blockIdx.x % BLOCKS_PER_PACK;
  const int wave  = tid >> 5;
  const int lane  = tid & 31;

  const v4f wv = *(const v4f*)(W + lane * 4);  // lane's 4 coefficients of W[0:128]

  const size_t nodebase = (size_t)b * NNODES +
                          (size_t)chunk * NODES_PER_BLOCK +
                          (size_t)wave * NODES_PER_WAVE;
  const float* nptr = nodes + nodebase * DLAT + (size_t)lane * 4;
  const int*   iptr = idx + nodebase;

#pragma unroll 4
  for (int j = 0; j < NODES_PER_WAVE; ++j) {
    const int g = __builtin_amdgcn_readfirstlane(iptr[j]);   // wave-uniform graph id
    const v4f v = *(const v4f*)(nptr + (size_t)j * DLAT);    // global_load_b128
    const float p = v.x * wv.x + v.y * wv.y + v.z * wv.z + v.w * wv.w;
    atomicAdd(&bins[g * 32 + lane], p);                      // ds_add_f32 (no rtn)
  }
  __syncthreads();

  if (tid < NGRAPH) {
    float s = 0.0f;
#pragma unroll
    for (int l = 0; l < 32; ++l) s += bins[tid * 32 + l];
    partials[blockIdx.x * NGRAPH + tid] = s;                 // deterministic, no atomics
  }
}

// ---------------------------------------------------------------------------
// Kernel B: WMMA readout of the global-latent half:
//   out[b*16+g] = sum_c partials[(b*32+c)*16+g]
//               + global_latent[b,g,:] · W[128:256] + bias
// The dot is computed as a 16x16x4 f32 WMMA chain: A = 16 (b,g) rows x K,
// B has W[128:256] in column 0 (other columns zero). One wave per 16 rows.
// A layout: lane = 16*h + m holds A[m][2h+v] in VGPR v.
// B layout: lane = 16*h + n holds B[2h+v][n] in VGPR v  -> col 0 lives in lanes 0,16.
// D layout: col 0 -> lane 0 has rows 0..7 in v[0..7], lane 16 has rows 8..15.
// W loads are UNCONDITIONAL (uniform per half-wave); the column-0 mask is
// applied as a value select (v_cndmask), so EXEC never changes -> no
// predicated-load branch regions around the WMMAs.
// ---------------------------------------------------------------------------
__global__ __launch_bounds__(32) void readout_wmma_kernel(
    const float* __restrict__ gl, const float* __restrict__ W,
    const float* __restrict__ bias, const float* __restrict__ partials,
    float* __restrict__ out) {
  const int lane = threadIdx.x;
  const int tile = blockIdx.x;      // == pack b (rows r = b*16 + g)
  const int half = lane >> 4;       // K-pair selector
  const int m    = lane & 15;       // row within tile (for A) / col (for B)
  const bool col0 = (m == 0);       // lanes 0 and 16 carry B column 0

  const float* arow = gl + ((size_t)tile * 16 + m) * DLAT + 2 * half;
  const float* wcol = W + DLAT + 2 * half;  // W[128:256]

  v8f acc = {};
#pragma unroll
  for (int k = 0; k < DLAT; k += 4) {
    v2f a;
    a.x = arow[k];
    a.y = arow[k + 1];
    // unconditional 8B load of the W pair, then mask by value (no EXEC change)
    const v2f wp = *(const v2f*)(wcol + k);
    v2f bb;
    bb.x = col0 ? wp.x : 0.0f;      // v_cndmask
    bb.y = col0 ? wp.y : 0.0f;      // v_cndmask
    acc = __builtin_amdgcn_wmma_f32_16x16x4_f32(
        /*neg_a=*/false, a, /*neg_b=*/false, bb,
        /*c_mod=*/(short)0, acc, /*reuse_a=*/false, /*reuse_b=*/false);
  }

  if (m == 0) {
    const int rbase = tile * 16 + half * 8;
    const float bv = bias[0];
#pragma unroll
    for (int i = 0; i < 8; ++i) {
      const int r = rbase + i;          // r = b*16 + g
      const int gb = r >> 4;            // pack
      const int gg = r & 15;            // graph
      float s = 0.0f;
      for (int c = 0; c < BLOCKS_PER_PACK; ++c)
        s += partials[(gb * BLOCKS_PER_PACK + c) * NGRAPH + gg];
      out[r] = acc[i] + s + bv;
    }
  }
}

// ---------------------------------------------------------------------------
// Inputs (setup_inputs order): 0=nodes f32, 1=global_latent f32, 2=W f32,
// 3=b f32 (1 elem), 4=node_graph_idx i32. Output: 256 f32.
// Scratch: 512 blocks * 16 graphs * 4B = 32 KiB of d_ws.
// ---------------------------------------------------------------------------
extern "C" void kernel_launch(void* const* d_in, const int* in_sizes, int n_in,
                              void* d_out, int out_size, void* d_ws, size_t ws_size,
                              hipStream_t stream) {
  const float* nodes = (const float*)d_in[0];
  const float* gl    = (const float*)d_in[1];
  const float* W     = (const float*)d_in[2];
  const float* bias  = (const float*)d_in[3];
  const int*   idx   = (const int*)d_in[4];
  float* out = (float*)d_out;
  float* partials = (float*)d_ws;  // [NPACK*BLOCKS_PER_PACK][NGRAPH]

  seg_dot_kernel<<<NPACK * BLOCKS_PER_PACK, THREADS_A, 0, stream>>>(
      nodes, W, idx, partials);
  readout_wmma_kernel<<<NPACK, 32, 0, stream>>>(gl, W, bias, partials, out);
}